// DynamicSpatialGNN_8495445311639
// MI455X (gfx1250) — compile-verified
//
#include <hip/hip_runtime.h>
#include <hip/hip_bf16.h>

typedef float v2f __attribute__((ext_vector_type(2)));
typedef float v8f __attribute__((ext_vector_type(8)));

#define B_   8
#define N_   32
#define T_   64
#define D_   128
#define E_   4
#define H_   32
#define W1LD (2*D_ + E_)   // 260
#define LD   132           // padded LDS row stride for 128-wide tiles
#define LS   33            // padded LDS row stride for 32-wide tiles

#if defined(__HIP_DEVICE_COMPILE__) && __has_builtin(__builtin_amdgcn_tensor_load_to_lds)
#define HAVE_TDM 1
typedef unsigned int u32x4 __attribute__((ext_vector_type(4)));
typedef int          i32x4 __attribute__((ext_vector_type(4)));
typedef int          i32x8 __attribute__((ext_vector_type(8)));
#else
#define HAVE_TDM 0
#endif

// ---- fp32 WMMA 16x16x4 wrapper -------------------------------------------
__device__ __forceinline__ v8f wmma4(v2f a, v2f b, v8f c) {
  return __builtin_amdgcn_wmma_f32_16x16x4_f32(
      /*neg_a=*/false, a, /*neg_b=*/false, b,
      /*c_mod=*/(short)0, c, /*reuse_a=*/false, /*reuse_b=*/false);
}

// A fragment (16x4) from LDS row-major tile: A[m][k]
__device__ __forceinline__ v2f frag_a_lds(const float* base, int row0, int k0, int ld) {
  int lane = threadIdx.x & 31;
  int m = lane & 15, h = lane >> 4;
  const float* p = base + (row0 + m) * ld + k0 + 2 * h;
  v2f a; a.x = p[0]; a.y = p[1];
  return a;
}

// B fragment (4x16) for X @ W^T with row-major W (rows = out dim, cols = K)
__device__ __forceinline__ v2f frag_b_wt_g(const float* __restrict__ W, int k0, int col0, int ldw) {
  int lane = threadIdx.x & 31;
  int j = col0 + (lane & 15), h = lane >> 4;
  const float2 p = *(const float2*)(W + (size_t)j * ldw + k0 + 2 * h);
  v2f b; b.x = p.x; b.y = p.y;
  return b;
}

// B fragment (4x16) where B[k][j] = M[j][k], M row-major in LDS (Q K^T)
__device__ __forceinline__ v2f frag_bt_lds(const float* base, int k0, int col0, int ld) {
  int lane = threadIdx.x & 31;
  int j = col0 + (lane & 15), h = lane >> 4;
  const float* p = base + j * ld + k0 + 2 * h;
  v2f b; b.x = p[0]; b.y = p[1];
  return b;
}

// B fragment (4x16) where B[k][n] = M[k][n], M row-major in LDS (alpha @ V)
__device__ __forceinline__ v2f frag_b_lds(const float* base, int k0, int col0, int ld) {
  int lane = threadIdx.x & 31;
  int n = col0 + (lane & 15), h = lane >> 4;
  v2f b;
  b.x = base[(k0 + 2 * h + 0) * ld + n];
  b.y = base[(k0 + 2 * h + 1) * ld + n];
  return b;
}

// C/D (16x16 f32, 8 VGPRs): reg r -> M = r + 8*(lane>>4), N = lane&15
__device__ __forceinline__ void store_c(float* base, v8f c, int row0, int col0, int ld) {
  int lane = threadIdx.x & 31;
  int n = lane & 15, h = lane >> 4;
  #pragma unroll
  for (int r = 0; r < 8; ++r)
    base[(row0 + r + 8 * h) * ld + col0 + n] = c[r];
}

__global__ __launch_bounds__(256, 2) void gnn_fused(
    const float* __restrict__ x, const float* __restrict__ edge,
    const float* __restrict__ Apr, const unsigned char* __restrict__ mask,
    const float* __restrict__ Wq, const float* __restrict__ Wk,
    const float* __restrict__ Wv, const float* __restrict__ Theta,
    const float* __restrict__ Wfuse, const float* __restrict__ W1,
    const float* __restrict__ b1, const float* __restrict__ W2,
    const float* __restrict__ b2, const float* __restrict__ gmm,
    const float* __restrict__ bet, const float* __restrict__ physw,
    const float* __restrict__ priow, float* __restrict__ out)
{
  __shared__ float Zs[N_ * LD];   // Z tile (kept for residual)
  __shared__ float Qs[N_ * LD];   // Q, later: spatial = alpha @ V
  __shared__ float Ks_[N_ * LD];  // K, later: O = spatial @ Theta^T
  __shared__ float Vs[N_ * LD];   // V
  __shared__ float hq_s[N_ * LS], hk_s[N_ * LS], lg[N_ * LS];
  __shared__ float w1e_s[H_ * E_], b1_s[H_], w2_s[H_], wf_s[8];
  __shared__ float gm_s[D_], bt_s[D_], sc[4];

  const int tid  = threadIdx.x;
  const int wave = tid >> 5, lane = tid & 31;
  const int bt = blockIdx.x;          // bt = b*T + t
  const int b  = bt >> 6, t = bt & 63;

  // ---- warm L2/WGP$ for the attention-phase inputs (consumed much later) --
  {
    const float* eb = edge + ((size_t)bt * N_ * N_) * E_;   // 4096 floats
    const float* pb = Apr  + ((size_t)bt * N_ * N_) * 5;    // 5120 floats
    if (tid < 128) __builtin_prefetch(eb + tid * 32, 0, 3);     // one per 128B line
    else           __builtin_prefetch(pb + (tid - 128) * 40, 0, 3);
  }

  // ---- stage constants ----
  for (int i = tid; i < H_ * E_; i += 256)
    w1e_s[i] = W1[(i >> 2) * W1LD + 2 * D_ + (i & 3)];
  if (tid < H_) { b1_s[tid] = b1[tid]; w2_s[tid] = W2[tid]; }
  if (tid < 5)  wf_s[tid] = Wfuse[tid];
  if (tid < D_) { gm_s[tid] = gmm[tid]; bt_s[tid] = bet[tid]; }
  if (tid == 0) { sc[0] = b2[0]; sc[1] = physw[0]; sc[2] = priow[0]; }

  // ---- stage Z tile: Z[n][d] = x[b][n][t][d] ----
#if HAVE_TDM
  if (wave == 0) {
    // Tensor Data Mover: 2D tile 128x32 f32, row stride T*D, with LDS
    // padding of 4 DWORDs after every 128 DWORDs (reproduces LD=132).
    unsigned lds_off = (unsigned)(unsigned long long)(const void*)Zs; // flat->LDS offset (addr[31:0])
    unsigned long long ga =
        (unsigned long long)(const void*)(x + ((size_t)b * N_ * T_ + t) * D_);
    u32x4 g0;
    g0[0] = 1u;                                            // count=1 valid user D#
    g0[1] = lds_off;                                       // lds_addr (bytes)
    g0[2] = (unsigned)(ga & 0xFFFFFFFFu);                  // global_addr[31:0]
    g0[3] = (unsigned)((ga >> 32) & 0x1FFFFFFu) | (2u << 30); // addr[56:32] | type=2
    i32x8 g1;
    g1[0] = (int)((2u << 16)            // data_size = 4 bytes
                | (1u << 20)            // pad_enable
                | (6u << 22)            // pad_interval: every 128 DWORDs
                | (3u << 25));          // pad_amount: 4 DWORDs
    g1[1] = (int)(128u << 16);          // tensor_dim0 = 128
    g1[2] = (int)(32u  << 16);          // tensor_dim1 = 32
    g1[3] = (int)(128u << 16);          // tile_dim0   = 128
    g1[4] = 32;                         // tile_dim1 = 32, tile_dim2 = 0
    g1[5] = T_ * D_;                    // tensor_dim0_stride = 8192
    g1[6] = 0;
    g1[7] = 0;
    i32x4 gz4 = {0, 0, 0, 0};
    i32x8 gz8 = {0, 0, 0, 0, 0, 0, 0, 0};
    __builtin_amdgcn_tensor_load_to_lds(g0, g1, gz4, gz4, gz8, 0);
    __builtin_amdgcn_s_wait_tensorcnt(0);
  }
#else
  for (int i = tid; i < N_ * D_ / 4; i += 256) {
    int flat = i * 4, n = flat >> 7, d = flat & (D_ - 1);
    const float4 v = *(const float4*)(x + ((((size_t)b * N_ + n) * T_ + t) * D_ + d));
    *(float4*)(&Zs[n * LD + d]) = v;
  }
#endif
  __syncthreads();

  // ---- Q, K, V = Z @ {Wq,Wk,Wv}^T : 48 C-tiles, 6 per wave ----
  #pragma unroll
  for (int i = 0; i < 6; ++i) {
    int tt = wave * 6 + i;
    int mat = tt >> 4, rem = tt & 15, rt = rem >> 3, ct = rem & 7;
    const float* Wm = (mat == 0) ? Wq : ((mat == 1) ? Wk : Wv);
    float* dst = (mat == 0) ? Qs : ((mat == 1) ? Ks_ : Vs);
    v8f acc = {};
    #pragma unroll 4
    for (int kk = 0; kk < D_ / 4; ++kk) {
      v2f a  = frag_a_lds(Zs, rt * 16, kk * 4, LD);
      v2f bb = frag_b_wt_g(Wm, kk * 4, ct * 16, D_);
      acc = wmma4(a, bb, acc);
    }
    store_c(dst, acc, rt * 16, ct * 16, LD);
  }
  __syncthreads();

  // ---- content = Q K^T ; hq = Q W1q^T ; hk = K W1k^T : 12 tiles ----
  for (int rep = 0; rep < 2; ++rep) {
    int tt = wave + rep * 8;
    if (tt < 12) {
      int which = tt >> 2, rt = (tt >> 1) & 1, ct = tt & 1;
      v8f acc = {};
      #pragma unroll 4
      for (int kk = 0; kk < D_ / 4; ++kk) {
        v2f a, bb;
        if (which == 0)      { a = frag_a_lds(Qs,  rt * 16, kk * 4, LD); bb = frag_bt_lds(Ks_, kk * 4, ct * 16, LD); }
        else if (which == 1) { a = frag_a_lds(Qs,  rt * 16, kk * 4, LD); bb = frag_b_wt_g(W1,      kk * 4, ct * 16, W1LD); }
        else                 { a = frag_a_lds(Ks_, rt * 16, kk * 4, LD); bb = frag_b_wt_g(W1 + D_, kk * 4, ct * 16, W1LD); }
        acc = wmma4(a, bb, acc);
      }
      store_c(which == 0 ? lg : (which == 1 ? hq_s : hk_s), acc, rt * 16, ct * 16, LS);
    }
  }
  __syncthreads();

  // ---- VALU epilogue: phys MLP + prior fuse + mask -> logits ----
  const float inv_sqrt_d = 0.08838834764831845f;  // 1/sqrt(128)
  for (int p = tid; p < N_ * N_; p += 256) {
    int n = p >> 5, m = p & 31;
    const float4 ev = *(const float4*)(edge + (((size_t)bt * N_ + n) * N_ + m) * E_);
    const float* pp = Apr + (((size_t)bt * N_ + n) * N_ + m) * 5;
    float pr = pp[0] * wf_s[0] + pp[1] * wf_s[1] + pp[2] * wf_s[2]
             + pp[3] * wf_s[3] + pp[4] * wf_s[4];
    if (!isfinite(pr)) pr = 0.0f;          // nan_to_num(nan/inf/-inf -> 0)
    pr = fmaxf(pr, 0.0f);                  // clip >= 0
    float plog = __logf(pr + 1e-6f);

    float acc = 0.0f;
    #pragma unroll
    for (int h = 0; h < H_; ++h) {
      float v = hq_s[n * LS + h] + hk_s[m * LS + h] + b1_s[h]
              + ev.x * w1e_s[h * 4 + 0] + ev.y * w1e_s[h * 4 + 1]
              + ev.z * w1e_s[h * 4 + 2] + ev.w * w1e_s[h * 4 + 3];
      acc = fmaf(fmaxf(v, 0.0f), w2_s[h], acc);
    }
    float logit = lg[n * LS + m] * inv_sqrt_d + sc[1] * (acc + sc[0]) + sc[2] * plog;
    bool inv = mask[b * N_ + n] || mask[b * N_ + m];
    lg[n * LS + m] = inv ? -1.0e9f : logit;
  }
  __syncthreads();

  // ---- softmax over m (row per wave32, lane = m) ----
  #pragma unroll
  for (int r = 0; r < 4; ++r) {
    int n = wave + r * 8;
    float v = lg[n * LS + lane];
    float mx = v;
    for (int off = 16; off; off >>= 1) mx = fmaxf(mx, __shfl_xor(mx, off));
    float e = __expf(v - mx);
    float s = e;
    for (int off = 16; off; off >>= 1) s += __shfl_xor(s, off);
    lg[n * LS + lane] = e / s;
  }
  __syncthreads();

  // ---- spatial = alpha @ V : 16 tiles (K = 32), 2 per wave -> Qs ----
  #pragma unroll
  for (int i = 0; i < 2; ++i) {
    int tt = wave * 2 + i, rt = tt >> 3, ct = tt & 7;
    v8f acc = {};
    #pragma unroll
    for (int kk = 0; kk < N_ / 4; ++kk) {
      v2f a  = frag_a_lds(lg, rt * 16, kk * 4, LS);
      v2f bb = frag_b_lds(Vs, kk * 4, ct * 16, LD);
      acc = wmma4(a, bb, acc);
    }
    store_c(Qs, acc, rt * 16, ct * 16, LD);
  }
  __syncthreads();

  // ---- O = spatial @ Theta^T : 16 tiles (K = 128), 2 per wave -> Ks_ ----
  #pragma unroll
  for (int i = 0; i < 2; ++i) {
    int tt = wave * 2 + i, rt = tt >> 3, ct = tt & 7;
    v8f acc = {};
    #pragma unroll 4
    for (int kk = 0; kk < D_ / 4; ++kk) {
      v2f a  = frag_a_lds(Qs, rt * 16, kk * 4, LD);
      v2f bb = frag_b_wt_g(Theta, kk * 4, ct * 16, D_);
      acc = wmma4(a, bb, acc);
    }
    store_c(Ks_, acc, rt * 16, ct * 16, LD);
  }
  __syncthreads();

  // ---- residual + layernorm + transposed store (b, n, t, d) ----
  #pragma unroll
  for (int r = 0; r < 4; ++r) {
    int n = wave + r * 8;
    float y[4], s = 0.0f, s2 = 0.0f;
    #pragma unroll
    for (int j = 0; j < 4; ++j) {
      int d = lane + j * 32;
      float v = Zs[n * LD + d] + Ks_[n * LD + d];
      y[j] = v; s += v; s2 += v * v;
    }
    for (int off = 16; off; off >>= 1) { s += __shfl_xor(s, off); s2 += __shfl_xor(s2, off); }
    float mu  = s * (1.0f / 128.0f);
    float var = s2 * (1.0f / 128.0f) - mu * mu;
    float rstd = rsqrtf(var + 1e-5f);
    bool mz = mask[b * N_ + n];
    float* op = out + (((size_t)(b * N_ + n)) * T_ + t) * D_;
    #pragma unroll
    for (int j = 0; j < 4; ++j) {
      int d = lane + j * 32;
      float o = (y[j] - mu) * rstd * gm_s[d] + bt_s[d];
      op[d] = mz ? 0.0f : o;
    }
  }
}

extern "C" void kernel_launch(void* const* d_in, const int* in_sizes, int n_in,
                              void* d_out, int out_size, void* d_ws, size_t ws_size,
                              hipStream_t stream) {
  (void)in_sizes; (void)n_in; (void)d_ws; (void)ws_size; (void)out_size;
  const float* x     = (const float*)d_in[0];
  const float* edge  = (const float*)d_in[1];
  const float* Apr   = (const float*)d_in[2];
  const unsigned char* msk = (const unsigned char*)d_in[3];  // jnp bool = 1 byte
  const float* Wq    = (const float*)d_in[4];
  const float* Wk    = (const float*)d_in[5];
  const float* Wv    = (const float*)d_in[6];
  const float* Theta = (const float*)d_in[7];
  const float* Wf    = (const float*)d_in[8];
  const float* W1    = (const float*)d_in[9];
  const float* b1    = (const float*)d_in[10];
  const float* W2    = (const float*)d_in[11];
  const float* b2    = (const float*)d_in[12];
  const float* gmm   = (const float*)d_in[13];
  const float* bet   = (const float*)d_in[14];
  const float* phw   = (const float*)d_in[15];
  const float* prw   = (const float*)d_in[16];
  float* out = (float*)d_out;

  gnn_fused<<<dim3(B_ * T_), dim3(256), 0, stream>>>(
      x, edge, Apr, msk, Wq, Wk, Wv, Theta, Wf, W1, b1, W2, b2,
      gmm, bet, phw, prw, out);
}